// PaddedHyperedgeAggregator_29643864277336
// MI455X (gfx1250) — compile-verified
//
#include <hip/hip_runtime.h>
#include <hip/hip_bf16.h>
#include <stdint.h>

#define NUM_NODES   100000
#define EMBED_DIM   256
#define MAX_SIZE    32
#define WAVES_PER_BLOCK 4
#define CHUNK_ROWS  8
#define ROW_BYTES   (EMBED_DIM * 4)   // 1024 B per embedding row

typedef uint32_t u32x4 __attribute__((ext_vector_type(4)));
typedef int      i32x4 __attribute__((ext_vector_type(4)));
typedef int      i32x8 __attribute__((ext_vector_type(8)));

#if defined(__has_builtin)
#if __has_builtin(__builtin_amdgcn_tensor_load_to_lds)
#define HAVE_TDM 1
#endif
#if __has_builtin(__builtin_amdgcn_s_wait_tensorcnt)
#define HAVE_TWAIT 1
#endif
#endif

static __device__ __forceinline__ void wait_tensorcnt_le(int n) {
#if defined(HAVE_TWAIT)
  if (n == 0) __builtin_amdgcn_s_wait_tensorcnt(0);
  else        __builtin_amdgcn_s_wait_tensorcnt(1);
#else
  if (n == 0) asm volatile("s_wait_tensorcnt 0x0" ::: "memory");
  else        asm volatile("s_wait_tensorcnt 0x1" ::: "memory");
#endif
}

#if defined(HAVE_TDM)
// Issue one TDM gather: up to 8 indexed rows of the [NUM_NODES, EMBED_DIM] f32
// tensor -> LDS at lds_off. Descriptor packing per CDNA5 ISA.
static __device__ __forceinline__ void tdm_gather8(uint64_t gbase, uint32_t lds_off,
                                                   int count,
                                                   int i0, int i1, int i2, int i3,
                                                   int i4, int i5, int i6, int i7) {
  u32x4 g0;
  // count=1 (valid), gather_index_size=1 (32-bit), gather_mode=1
  g0.x = 0xC0000001u;
  g0.y = lds_off;                                   // lds_addr (bytes)
  g0.z = (uint32_t)gbase;                           // global_addr[31:0]
  g0.w = ((uint32_t)(gbase >> 32) & 0x01FFFFFFu)    // global_addr[56:32]
         | 0x80000000u;                             // type = 2 ("image")
  i32x8 g1;
  g1[0] = (int)(2u << 16);                          // wg_mask=0, data_size=2 (4B)
  g1[1] = (int)((uint32_t)(EMBED_DIM & 0xFFFF) << 16);   // tensor_dim0[15:0]
  g1[2] = (int)(((uint32_t)NUM_NODES & 0xFFFFu) << 16);  // dim0 hi16=0 | tensor_dim1 lo16
  g1[3] = (int)(((uint32_t)NUM_NODES >> 16) | ((uint32_t)EMBED_DIM << 16)); // dim1 hi | tile_dim0
  g1[4] = count;                                    // tile_dim1 = #valid indices
  g1[5] = EMBED_DIM;                                // tensor_dim0_stride (elements)
  g1[6] = 0;
  g1[7] = 0;
  i32x4 g2; g2[0] = i0; g2[1] = i1; g2[2] = i2; g2[3] = i3;   // row_index_0..3
  i32x4 g3; g3[0] = i4; g3[1] = i5; g3[2] = i6; g3[3] = i7;   // row_index_4..7
#if defined(__clang_major__) && (__clang_major__ >= 23)
  i32x8 z8 = {0, 0, 0, 0, 0, 0, 0, 0};
  __builtin_amdgcn_tensor_load_to_lds(g0, g1, g2, g3, z8, 0);
#else
  __builtin_amdgcn_tensor_load_to_lds(g0, g1, g2, g3, 0);
#endif
}
#endif // HAVE_TDM

static __device__ __forceinline__ void acc4(float4& a, const float4 b) {
  a.x += b.x; a.y += b.y; a.z += b.z; a.w += b.w;
}

__global__ void __launch_bounds__(WAVES_PER_BLOCK * 32)
padded_hyperedge_mean_kernel(const float* __restrict__ emb,
                             const int* __restrict__ edges,    // [B, 32]
                             const int* __restrict__ lengths,  // [B]
                             float* __restrict__ out,          // [B, 256]
                             int batch) {
#if defined(HAVE_TDM)
  __shared__ float stage[WAVES_PER_BLOCK][2][CHUNK_ROWS][EMBED_DIM]; // 64 KB
#endif
  const int lane = threadIdx.x & 31;
  const int wave = threadIdx.x >> 5;
  const int e = blockIdx.x * WAVES_PER_BLOCK + wave;
  if (e >= batch) return;

  const int L = lengths[e];                          // in [1, 32]
  const int myIdx = edges[e * MAX_SIZE + lane];      // lane s holds idx[e, s]

  float4 acc0 = make_float4(0.f, 0.f, 0.f, 0.f);
  float4 acc1 = make_float4(0.f, 0.f, 0.f, 0.f);

#if defined(HAVE_TDM)
  const uint64_t gbase = (uint64_t)(uintptr_t)emb;
  const uint32_t ldsBase = (uint32_t)(wave * (2 * CHUNK_ROWS * ROW_BYTES));
  const int nch = (L + CHUNK_ROWS - 1) / CHUNK_ROWS;

  auto issue = [&](int c) {
    const int base = c * CHUNK_ROWS;
    int ix[8];
#pragma unroll
    for (int r = 0; r < 8; ++r) {
      int v = __shfl(myIdx, base + r, 32);
      ix[r] = __builtin_amdgcn_readfirstlane(v);     // descriptors live in SGPRs
    }
    int cnt = L - base; if (cnt > CHUNK_ROWS) cnt = CHUNK_ROWS;
    tdm_gather8(gbase, ldsBase + (uint32_t)((c & 1) * CHUNK_ROWS * ROW_BYTES),
                cnt, ix[0], ix[1], ix[2], ix[3], ix[4], ix[5], ix[6], ix[7]);
  };

  issue(0);
  for (int c = 0; c < nch; ++c) {
    if (c + 1 < nch) {                 // prefetch next chunk, keep one in flight
      issue(c + 1);
      wait_tensorcnt_le(1);
    } else {
      wait_tensorcnt_le(0);
    }
    int v = L - c * CHUNK_ROWS; if (v > CHUNK_ROWS) v = CHUNK_ROWS;
    const float* buf = &stage[wave][c & 1][0][0];
    for (int r = 0; r < v; ++r) {
      const float4* p = (const float4*)(buf + r * EMBED_DIM + lane * 8);
      acc4(acc0, p[0]);                // ds_load_b128 x2: 32 lanes x 32B = row
      acc4(acc1, p[1]);
    }
  }
#else
  // Fallback: direct coalesced gather through L2 (table is L2-resident).
  for (int s = 0; s < L; ++s) {
    int ri = __shfl(myIdx, s, 32);
    const float4* p = (const float4*)(emb + (size_t)ri * EMBED_DIM + lane * 8);
    acc4(acc0, p[0]);
    acc4(acc1, p[1]);
  }
#endif

  const float inv = 1.0f / (float)L;
  float* o = out + (size_t)e * EMBED_DIM + lane * 8;
  float4 o0 = make_float4(acc0.x * inv, acc0.y * inv, acc0.z * inv, acc0.w * inv);
  float4 o1 = make_float4(acc1.x * inv, acc1.y * inv, acc1.z * inv, acc1.w * inv);
  ((float4*)o)[0] = o0;
  ((float4*)o)[1] = o1;
}

extern "C" void kernel_launch(void* const* d_in, const int* in_sizes, int n_in,
                              void* d_out, int out_size, void* d_ws, size_t ws_size,
                              hipStream_t stream) {
  const float* emb     = (const float*)d_in[0];
  const int*   edges   = (const int*)d_in[1];
  const int*   lengths = (const int*)d_in[2];
  float*       out     = (float*)d_out;
  const int batch = in_sizes[2];   // lengths has one entry per hyperedge

  const int threads = WAVES_PER_BLOCK * 32;
  const int blocks  = (batch + WAVES_PER_BLOCK - 1) / WAVES_PER_BLOCK;
  padded_hyperedge_mean_kernel<<<blocks, threads, 0, stream>>>(
      emb, edges, lengths, out, batch);
}